// XConv_24584392802802
// MI455X (gfx1250) — compile-verified
//
#include <hip/hip_runtime.h>

// Fused XConv for MI455X (gfx1250, wave32).
// Block = 8 waves x 2 points = 16 points. Per-point 16x16 contractions use
// f32 WMMA 16x16x4; the two per-point matvecs (wconv, wpw) are batched into
// block-level WMMA GEMMs across the 16 points to amortize weight traffic.

typedef __attribute__((ext_vector_type(2))) float v2f;
typedef __attribute__((ext_vector_type(8))) float v8f;

#define WAVES 8
#define PTS_PER_BLOCK 16
#define NPOINTS (32 * 1024)    // N*P

// ---- dynamic LDS layout (float offsets) ----
constexpr int F_W2   = 0;                    // w2 staged 64 x 65
constexpr int F_PL   = F_W2 + 64 * 65;       // pts_local, 16 pts x 48
constexpr int F_X0   = F_PL + 16 * 48;       // X0, 16 pts x 260 (256 pad 260)
constexpr int F_TMP  = F_X0 + 16 * 260;      // tmp, 16 pts x 264 (256 pad 264)
constexpr int F_WAVE = F_TMP + 16 * 264;     // per-wave scratch below
constexpr int W_CAT  = 0;                    // fts_cat 16 x 132
constexpr int W_FX   = 16 * 132;             // h1 / fts_X 16 x 132
constexpr int W_XA   = 2 * 16 * 132;         // X stage a 16 x 17
constexpr int W_XB   = W_XA + 16 * 17;       // X stage b 16 x 17
constexpr int WAVE_FLOATS = 2 * 16 * 132 + 2 * 16 * 17;   // 4768
constexpr int SMEM_FLOATS = F_WAVE + WAVES * WAVE_FLOATS; // 51456
constexpr int SMEM_BYTES  = SMEM_FLOATS * 4;              // 205824 (< 320KB/WGP)

#if __has_builtin(__builtin_amdgcn_global_load_async_to_lds_b128)
#define HAVE_ASYNC_LDS 1
typedef int vs4i __attribute__((vector_size(16)));
typedef __attribute__((address_space(1))) vs4i gvs4i;   // global int4
typedef __attribute__((address_space(3))) vs4i lvs4i;   // LDS int4
#endif

__device__ __forceinline__ float elu_f(float x) {
    return x > 0.0f ? x : (__expf(x) - 1.0f);
}

__device__ __forceinline__ v8f wmma_f32_k4(v2f a, v2f b, v8f c) {
    return __builtin_amdgcn_wmma_f32_16x16x4_f32(
        false, a, false, b, (short)0, c, false, false);
}

__global__ __launch_bounds__(32 * WAVES)
void xconv_fused(const float* __restrict__ rep_pt,
                 const float* __restrict__ pts,
                 const float* __restrict__ fts,
                 const float* __restrict__ w1,    const float* __restrict__ b1,
                 const float* __restrict__ w2,    const float* __restrict__ b2,
                 const float* __restrict__ wconv, const float* __restrict__ bconv,
                 const float* __restrict__ wdep1, const float* __restrict__ bdep1,
                 const float* __restrict__ wdep2, const float* __restrict__ bdep2,
                 const float* __restrict__ wdw,   const float* __restrict__ bdw,
                 const float* __restrict__ wpw,   const float* __restrict__ bpw,
                 float* __restrict__ out)
{
    extern __shared__ float smem[];

    const int tid  = threadIdx.x;
    const int wv   = tid >> 5;
    const int lane = tid & 31;
    const int row  = lane & 15;
    const int half = lane >> 4;
    const int kk0  = 2 * half;          // K-pair base inside a K=4 chunk

    const int ptBase = blockIdx.x * PTS_PER_BLOCK;

    float* s_w2  = smem + F_W2;
    float* s_pl  = smem + F_PL;          // [pt][48]
    float* s_x0  = smem + F_X0;          // [pt][260]
    float* s_tm  = smem + F_TMP;         // [pt][264]
    float* s_cat = smem + F_WAVE + wv * WAVE_FLOATS + W_CAT;  // [16][132]
    float* s_fx  = smem + F_WAVE + wv * WAVE_FLOATS + W_FX;   // [16][132]
    float* s_xa  = smem + F_WAVE + wv * WAVE_FLOATS + W_XA;   // [16][17]
    float* s_xb  = smem + F_WAVE + wv * WAVE_FLOATS + W_XB;   // [16][17]

    // ---------------- Stage: w2 (coop) + pts_local (all 16 points) --------
    for (int i = tid; i < 64 * 64; i += 32 * WAVES)
        s_w2[(i >> 6) * 65 + (i & 63)] = w2[i];

    {   // each wave stages its 2 points; half selects the point
        int ptl = wv * 2 + half;
        int pt  = ptBase + ptl;
        #pragma unroll
        for (int d = 0; d < 3; ++d)
            s_pl[ptl * 48 + row * 3 + d] =
                pts[pt * 48 + row * 3 + d] - rep_pt[pt * 3 + d];
    }
    __syncthreads();

    // ------- Block GEMM 1: X0(16pts x 256) = P(16x48) @ wconvT(48x256) ----
    // K index u = k*3 + d  ->  wconvT[u][o] = wconv[o*48 + (u%3)*16 + u/3]
    #pragma unroll
    for (int tt = 0; tt < 2; ++tt) {
        const int t = wv + 8 * tt;            // 16 N-tiles over 8 waves
        const int o = t * 16 + row;
        v8f acc0 = {}, acc1 = {};
        #pragma unroll
        for (int j = 0; j < 12; ++j) {
            int u0 = 4 * j + kk0, u1 = u0 + 1;
            v2f a, b;
            a.x = s_pl[row * 48 + u0];        // row = point index
            a.y = s_pl[row * 48 + u1];
            b.x = wconv[o * 48 + (u0 % 3) * 16 + (u0 / 3)];
            b.y = wconv[o * 48 + (u1 % 3) * 16 + (u1 / 3)];
            if (j & 1) acc1 = wmma_f32_k4(a, b, acc1);
            else       acc0 = wmma_f32_k4(a, b, acc0);
        }
        v8f acc = acc0 + acc1;
        float bias = bconv[o];
        #pragma unroll
        for (int v = 0; v < 8; ++v)
            s_x0[(v + 8 * half) * 260 + o] = elu_f(acc[v] + bias);
    }
    __syncthreads();

    // =================== per-point pipeline (2 points/wave) ===============
    for (int it = 0; it < 2; ++it) {
        const int ptl = wv * 2 + it;
        const int pt  = ptBase + ptl;

        // stage fts (16x64 contiguous) -> s_cat cols 64..127
        {
            const float* gsrc = fts + pt * 1024;
#ifdef HAVE_ASYNC_LDS
            #pragma unroll
            for (int g = 0; g < 8; ++g) {
                int idx = lane + 32 * g;                 // k = 2g+half, c4 = row
                const float* gp = gsrc + idx * 4;
                float* lp = s_cat + (2 * g + half) * 132 + 64 + row * 4;
                __builtin_amdgcn_global_load_async_to_lds_b128(
                    (gvs4i*)gp, (lvs4i*)lp, 0, 0);
            }
#else
            #pragma unroll
            for (int g = 0; g < 8; ++g) {
                int idx = lane + 32 * g;
                float4 v = *(const float4*)(gsrc + idx * 4);
                *(float4*)&s_cat[(2 * g + half) * 132 + 64 + row * 4] = v;
            }
#endif
        }

        // Phase 1: h1 = elu(ptsl @ w1 + b1) (16x64) -> s_fx   (K 3 padded 4)
        {
            v2f a;
            a.x = s_pl[ptl * 48 + row * 3 + kk0];            // d = kk0 (0 or 2)
            a.y = (half == 0) ? s_pl[ptl * 48 + row * 3 + 1] : 0.0f;
            #pragma unroll
            for (int t = 0; t < 4; ++t) {
                v2f b;
                b.x = (kk0 + 0 < 3) ? w1[(kk0 + 0) * 64 + t * 16 + row] : 0.0f;
                b.y = (kk0 + 1 < 3) ? w1[(kk0 + 1) * 64 + t * 16 + row] : 0.0f;
                v8f acc = {};
                acc = wmma_f32_k4(a, b, acc);
                float bias = b1[t * 16 + row];
                #pragma unroll
                for (int v = 0; v < 8; ++v)
                    s_fx[(v + 8 * half) * 132 + t * 16 + row] = elu_f(acc[v] + bias);
            }
        }
        __syncthreads();

        // Phase 2: h2 = elu(h1 @ w2 + b2) (16x64) -> s_cat cols 0..63
        {
            v2f a[16];
            #pragma unroll
            for (int j = 0; j < 16; ++j) {
                a[j].x = s_fx[row * 132 + 4 * j + kk0 + 0];
                a[j].y = s_fx[row * 132 + 4 * j + kk0 + 1];
            }
            #pragma unroll
            for (int t = 0; t < 4; ++t) {
                v8f acc0 = {}, acc1 = {};
                #pragma unroll
                for (int j = 0; j < 16; ++j) {
                    v2f b;
                    b.x = s_w2[(4 * j + kk0 + 0) * 65 + t * 16 + row];
                    b.y = s_w2[(4 * j + kk0 + 1) * 65 + t * 16 + row];
                    if (j & 1) acc1 = wmma_f32_k4(a[j], b, acc1);
                    else       acc0 = wmma_f32_k4(a[j], b, acc0);
                }
                v8f acc = acc0 + acc1;
                float bias = b2[t * 16 + row];
                #pragma unroll
                for (int v = 0; v < 8; ++v)
                    s_cat[(v + 8 * half) * 132 + t * 16 + row] = elu_f(acc[v] + bias);
            }
        }
        __syncthreads();

        // Phase 4: X1[i][j] = elu(sum_k X0[k][i]*wdep1[i][j][k] + b) -> s_xb
        #pragma unroll
        for (int q = 0; q < 8; ++q) {
            int o = lane + 32 * q;           // o = i*16 + j
            int i = o >> 4;
            float acc = bdep1[o];
            const float* wr = wdep1 + o * 16;
            #pragma unroll
            for (int k = 0; k < 16; ++k) acc += s_x0[ptl * 260 + k * 16 + i] * wr[k];
            s_xb[o + (o >> 4)] = elu_f(acc);  // [i*17 + j]
        }
        __syncthreads();

        // Phase 5: X2[i][j] = sum_k X1[k][i]*wdep2[i][j][k] + b  -> s_xa
        #pragma unroll
        for (int q = 0; q < 8; ++q) {
            int o = lane + 32 * q;
            int i = o >> 4;
            float acc = bdep2[o];
            const float* wr = wdep2 + o * 16;
            #pragma unroll
            for (int k = 0; k < 16; ++k) acc += s_xb[k * 17 + i] * wr[k];
            s_xa[o + (o >> 4)] = acc;
        }
        __syncthreads();

#ifdef HAVE_ASYNC_LDS
        asm volatile("s_wait_asynccnt 0" ::: "memory");   // fts staged
#endif

        // Phase 6: fts_X = X2(16x16) @ fts_cat(16x128) via WMMA -> s_fx
        {
            v2f a[4];
            #pragma unroll
            for (int j4 = 0; j4 < 4; ++j4) {
                a[j4].x = s_xa[row * 17 + 4 * j4 + kk0 + 0];
                a[j4].y = s_xa[row * 17 + 4 * j4 + kk0 + 1];
            }
            #pragma unroll
            for (int t = 0; t < 8; ++t) {
                v8f acc = {};
                #pragma unroll
                for (int j4 = 0; j4 < 4; ++j4) {
                    v2f b;
                    b.x = s_cat[(4 * j4 + kk0 + 0) * 132 + t * 16 + row];
                    b.y = s_cat[(4 * j4 + kk0 + 1) * 132 + t * 16 + row];
                    acc = wmma_f32_k4(a[j4], b, acc);
                }
                #pragma unroll
                for (int v = 0; v < 8; ++v)
                    s_fx[(v + 8 * half) * 132 + t * 16 + row] = acc[v];
            }
        }
        __syncthreads();

        // Phase 7: tmp[c][m] = sum_k ftsX[k][c]*wdw[c][m][k] + bdw -> s_tm[ptl]
        #pragma unroll
        for (int q = 0; q < 8; ++q) {
            int idx = lane + 32 * q;         // idx = c*2 + m
            int c = idx >> 1;
            float acc = bdw[idx];
            const float* wr = wdw + idx * 16;
            #pragma unroll
            for (int k = 0; k < 16; ++k) acc += s_fx[k * 132 + c] * wr[k];
            s_tm[ptl * 264 + idx] = acc;
        }
        __syncthreads();
    }

    // ------- Block GEMM 2: out(16pts x 128) = tmp(16x256) @ wpwT(256x128) --
    // wpwT[cm][o] = wpw[o*256 + cm]; wave wv owns N-tile t = wv.
    {
        const int t = wv;
        const int o = t * 16 + row;
        v8f acc0 = {}, acc1 = {};
        #pragma unroll
        for (int j = 0; j < 64; ++j) {
            int u0 = 4 * j + kk0;
            v2f a, b;
            a.x = s_tm[row * 264 + u0];      // row = point index
            a.y = s_tm[row * 264 + u0 + 1];
            float2 wb = *(const float2*)(wpw + o * 256 + u0);
            b.x = wb.x;
            b.y = wb.y;
            if (j & 1) acc1 = wmma_f32_k4(a, b, acc1);
            else       acc0 = wmma_f32_k4(a, b, acc0);
        }
        v8f acc = acc0 + acc1;
        float bias = bpw[o];
        #pragma unroll
        for (int v = 0; v < 8; ++v) {
            int p = ptBase + v + 8 * half;   // D row = point
            out[p * 128 + o] = elu_f(acc[v] + bias);
        }
    }
}

extern "C" void kernel_launch(void* const* d_in, const int* in_sizes, int n_in,
                              void* d_out, int out_size, void* d_ws, size_t ws_size,
                              hipStream_t stream) {
    (void)in_sizes; (void)n_in; (void)out_size; (void)d_ws; (void)ws_size;
    const float* rep   = (const float*)d_in[0];
    const float* pts   = (const float*)d_in[1];
    const float* fts   = (const float*)d_in[2];
    const float* w1    = (const float*)d_in[3];
    const float* b1    = (const float*)d_in[4];
    const float* w2    = (const float*)d_in[5];
    const float* b2    = (const float*)d_in[6];
    const float* wconv = (const float*)d_in[7];
    const float* bconv = (const float*)d_in[8];
    const float* wdep1 = (const float*)d_in[9];
    const float* bdep1 = (const float*)d_in[10];
    const float* wdep2 = (const float*)d_in[11];
    const float* bdep2 = (const float*)d_in[12];
    const float* wdw   = (const float*)d_in[13];
    const float* bdw   = (const float*)d_in[14];
    const float* wpw   = (const float*)d_in[15];
    const float* bpw   = (const float*)d_in[16];

    static int attr_done = 0;
    if (!attr_done) {
        (void)hipFuncSetAttribute((const void*)xconv_fused,
                                  hipFuncAttributeMaxDynamicSharedMemorySize,
                                  SMEM_BYTES);
        attr_done = 1;
    }

    dim3 grid(NPOINTS / PTS_PER_BLOCK), block(32 * WAVES);
    hipLaunchKernelGGL(xconv_fused, grid, block, SMEM_BYTES, stream,
                       rep, pts, fts, w1, b1, w2, b2, wconv, bconv,
                       wdep1, bdep1, wdep2, bdep2, wdw, bdw, wpw, bpw,
                       (float*)d_out);
}